// SkipGCN3Layer_44212393345739
// MI455X (gfx1250) — compile-verified
//
#include <hip/hip_runtime.h>

typedef float v2f __attribute__((ext_vector_type(2)));
typedef float v8f __attribute__((ext_vector_type(8)));

#define N_FEAT_IN 128

// ---------------- degree / normalization ----------------

__global__ void deg_init_kernel(float* __restrict__ deg, int N) {
    int i = blockIdx.x * blockDim.x + threadIdx.x;
    if (i < N) deg[i] = 1.0f;  // self-loop
}

__device__ __forceinline__ void atomic_add_f32(float* p, float v) {
    unsafeAtomicAdd(p, v);  // native global_atomic_add_f32 on gfx1250
}

__global__ void deg_acc_kernel(const long long* __restrict__ dst,
                               float* __restrict__ deg, long long E) {
    long long e = (long long)blockIdx.x * blockDim.x + threadIdx.x;
    if (e < E) atomic_add_f32(&deg[dst[e]], 1.0f);
}

__global__ void rsqrt_kernel(float* __restrict__ deg, int N) {
    int i = blockIdx.x * blockDim.x + threadIdx.x;
    if (i < N) deg[i] = rsqrtf(deg[i]);  // in place -> dinv
}

// ---------------- WMMA f32 GEMM: H[N,F] = X[N,K] @ W[K,F] ----------------
// One wave32 per 16x16 output tile. Requires N % 16 == 0 (N = 100000 here).
// A (16x4 f32): lane l -> row l%16, K-pair base 2*(l/16). B (4x16) mirrored.
// C/D (16x16 f32, 8 VGPRs): VGPR r -> row r + 8*(l/16), col l%16.

template <int K, int F>
__global__ __launch_bounds__(128) void gemm16_wmma(const float* __restrict__ X,
                                                   const float* __restrict__ Wm,
                                                   float* __restrict__ H, int N) {
    constexpr int TN = F / 16;
    int wave = (int)((blockIdx.x * (long long)blockDim.x + threadIdx.x) >> 5);
    int lane = threadIdx.x & 31;
    int tiles_m = N >> 4;
    int tile_m = wave / TN;
    int tile_n = wave % TN;
    if (tile_m >= tiles_m) return;  // wave-uniform: EXEC stays all-ones for WMMA

    int m  = lane & 15;
    int kb = (lane >> 4) << 1;                 // 0 or 2
    const float* xrow = X + (size_t)(tile_m * 16 + m) * K;
    int col = tile_n * 16 + m;

    v8f c = {};
#pragma unroll
    for (int k = 0; k < K; k += 4) {
        v2f a = *(const v2f*)(xrow + k + kb);  // 8B-aligned: k+kb even, K even
        v2f b;
        b.x = Wm[(size_t)(k + kb) * F + col];
        b.y = Wm[(size_t)(k + kb + 1) * F + col];
        c = __builtin_amdgcn_wmma_f32_16x16x4_f32(false, a, false, b,
                                                  (short)0, c, false, false);
    }

    int rbase = tile_m * 16 + ((lane >> 4) << 3);
    int cidx  = tile_n * 16 + (lane & 15);
#pragma unroll
    for (int r = 0; r < 8; ++r)
        H[(size_t)(rbase + r) * F + cidx] = c[r];
}

// ---------------- self-loop: agg = dinv^2 * h (full overwrite) ----------------

template <int F>
__global__ void self_init_kernel(const float* __restrict__ dinv,
                                 const float* __restrict__ h,
                                 float* __restrict__ agg, long long NF) {
    long long t = (long long)blockIdx.x * blockDim.x + threadIdx.x;
    if (t >= NF) return;
    float di = dinv[t / F];
    agg[t] = di * di * h[t];
}

// ---------------- edge scatter: agg[dst] += dinv[s]*dinv[d]*h[src] ----------------
// Thread handles a float4 chunk of one edge; consecutive lanes -> consecutive
// chunks of the same row (coalesced gather + coalesced atomics).

template <int F>
__global__ void gcn_scatter_kernel(const long long* __restrict__ src,
                                   const long long* __restrict__ dst,
                                   const float* __restrict__ dinv,
                                   const float* __restrict__ h,
                                   float* __restrict__ agg, long long E) {
    constexpr int C = F / 4;
    long long t = (long long)blockIdx.x * blockDim.x + threadIdx.x;
    long long e = t / C;
    int ch = (int)(t % C);
    if (e >= E) return;
    long long s = src[e], d = dst[e];
    float nrm = dinv[s] * dinv[d];
    float4 v = *((const float4*)(h + (size_t)s * F) + ch);
    float* ap = agg + (size_t)d * F + (size_t)ch * 4;
    atomic_add_f32(ap + 0, nrm * v.x);
    atomic_add_f32(ap + 1, nrm * v.y);
    atomic_add_f32(ap + 2, nrm * v.z);
    atomic_add_f32(ap + 3, nrm * v.w);
}

// ---------------- finalize: out = relu(agg + b) [+ s1 + sb1] [+ s2 + sb2] ----------------

template <int F>
__global__ void finalize_kernel(const float* __restrict__ agg,
                                const float* __restrict__ bias,
                                const float* __restrict__ s1, const float* __restrict__ sb1,
                                const float* __restrict__ s2, const float* __restrict__ sb2,
                                float* __restrict__ out, long long NF) {
    long long t = (long long)blockIdx.x * blockDim.x + threadIdx.x;
    if (t >= NF) return;
    int f = (int)(t % F);
    float v = fmaxf(agg[t] + bias[f], 0.0f);
    if (s1) v += s1[t] + sb1[f];
    if (s2) v += s2[t] + sb2[f];
    out[t] = v;
}

// ---------------- host-side orchestration ----------------

static inline unsigned grid1(long long n, int b) { return (unsigned)((n + b - 1) / b); }

extern "C" void kernel_launch(void* const* d_in, const int* in_sizes, int n_in,
                              void* d_out, int out_size, void* d_ws, size_t ws_size,
                              hipStream_t stream) {
    const float*     x    = (const float*)d_in[0];
    const long long* ei   = (const long long*)d_in[1];
    const float* W1   = (const float*)d_in[2];
    const float* b1   = (const float*)d_in[3];
    const float* W2   = (const float*)d_in[4];
    const float* b2   = (const float*)d_in[5];
    const float* W3   = (const float*)d_in[6];
    const float* b3   = (const float*)d_in[7];
    const float* Ws02 = (const float*)d_in[8];
    const float* bs02 = (const float*)d_in[9];
    const float* Ws03 = (const float*)d_in[10];
    const float* bs03 = (const float*)d_in[11];
    const float* Ws13 = (const float*)d_in[12];
    const float* bs13 = (const float*)d_in[13];

    const int       N = in_sizes[0] / N_FEAT_IN;   // 100000
    const long long E = in_sizes[1] / 2;           // 3200000
    const long long* srcp = ei;
    const long long* dstp = ei + E;

    // workspace layout (floats, 1KB-aligned slices)
    float* ws = (float*)d_ws;
    auto al = [](size_t v) { return (v + 255) & ~(size_t)255; };
    size_t o = 0;
    float* dinv = ws + o; o += al((size_t)N);
    float* h1   = ws + o; o += al((size_t)N * 64);   // later reused as x1
    float* agg1 = ws + o; o += al((size_t)N * 64);
    float* h2   = ws + o; o += al((size_t)N * 32);   // later reused as x2
    float* agg2 = ws + o; o += al((size_t)N * 32);
    float* s02  = ws + o; o += al((size_t)N * 32);
    float* h3   = ws + o; o += al((size_t)N * 16);
    float* agg3 = ws + o; o += al((size_t)N * 16);
    float* s03  = ws + o; o += al((size_t)N * 16);
    float* s13  = ws + o; o += al((size_t)N * 16);
    float* x1 = h1;
    float* x2 = h2;
    float* outp = (float*)d_out;

    const int B = 256;

    // degree + dinv
    deg_init_kernel<<<grid1(N, B), B, 0, stream>>>(dinv, N);
    deg_acc_kernel<<<grid1(E, B), B, 0, stream>>>(dstp, dinv, E);
    rsqrt_kernel<<<grid1(N, B), B, 0, stream>>>(dinv, N);

    auto gemm_grid = [&](int F_) { return grid1((long long)(N / 16) * (F_ / 16) * 32, 128); };

    // ---- layer 1: x1 = relu(gcn(x, W1, b1)) ----
    gemm16_wmma<128, 64><<<gemm_grid(64), 128, 0, stream>>>(x, W1, h1, N);
    self_init_kernel<64><<<grid1((long long)N * 64, B), B, 0, stream>>>(dinv, h1, agg1, (long long)N * 64);
    gcn_scatter_kernel<64><<<grid1(E * 16, B), B, 0, stream>>>(srcp, dstp, dinv, h1, agg1, E);
    finalize_kernel<64><<<grid1((long long)N * 64, B), B, 0, stream>>>(
        agg1, b1, nullptr, nullptr, nullptr, nullptr, x1, (long long)N * 64);

    // ---- layer 2: x2 = relu(gcn(x1, W2, b2)) + x@Ws02 + bs02 ----
    gemm16_wmma<64, 32><<<gemm_grid(32), 128, 0, stream>>>(x1, W2, h2, N);
    gemm16_wmma<128, 32><<<gemm_grid(32), 128, 0, stream>>>(x, Ws02, s02, N);
    self_init_kernel<32><<<grid1((long long)N * 32, B), B, 0, stream>>>(dinv, h2, agg2, (long long)N * 32);
    gcn_scatter_kernel<32><<<grid1(E * 8, B), B, 0, stream>>>(srcp, dstp, dinv, h2, agg2, E);
    finalize_kernel<32><<<grid1((long long)N * 32, B), B, 0, stream>>>(
        agg2, b2, s02, bs02, nullptr, nullptr, x2, (long long)N * 32);

    // ---- layer 3: out = relu(gcn(x2, W3, b3)) + x@Ws03 + bs03 + x1@Ws13 + bs13 ----
    gemm16_wmma<32, 16><<<gemm_grid(16), 128, 0, stream>>>(x2, W3, h3, N);
    gemm16_wmma<128, 16><<<gemm_grid(16), 128, 0, stream>>>(x, Ws03, s03, N);
    gemm16_wmma<64, 16><<<gemm_grid(16), 128, 0, stream>>>(x1, Ws13, s13, N);
    self_init_kernel<16><<<grid1((long long)N * 16, B), B, 0, stream>>>(dinv, h3, agg3, (long long)N * 16);
    gcn_scatter_kernel<16><<<grid1(E * 4, B), B, 0, stream>>>(srcp, dstp, dinv, h3, agg3, E);
    finalize_kernel<16><<<grid1((long long)N * 16, B), B, 0, stream>>>(
        agg3, b3, s03, bs03, s13, bs13, outp, (long long)N * 16);
}